// ExtendedKalmanFilter_7310034338113
// MI455X (gfx1250) — compile-verified
//
#include <hip/hip_runtime.h>

#define BATCH 8192
#define NDIM 64
#define MDIM 32
#define EPSV 1e-6f

typedef __attribute__((ext_vector_type(2))) float v2f;
typedef __attribute__((ext_vector_type(8))) float v8f;
typedef __attribute__((ext_vector_type(4))) int   v4i;

// ---------------------------------------------------------------------------
// Async global->LDS staging (CDNA5 GLOBAL_LOAD_ASYNC_TO_LDS_B128, ASYNCcnt).
// Builtin signature (from clang diagnostic): (int4 AS1*, int4 AS3*, imm, imm).
// Guarded: falls back to VGPR-staged float4 copies if builtins are absent.
// ---------------------------------------------------------------------------
#if __has_builtin(__builtin_amdgcn_global_load_async_to_lds_b128) && \
    __has_builtin(__builtin_amdgcn_s_wait_asynccnt)
#define HAVE_ASYNC_LDS 1
#endif

#ifdef HAVE_ASYNC_LDS
typedef __attribute__((address_space(1))) v4i gv4i;   // global (prints as __device__)
typedef __attribute__((address_space(3))) v4i lv4i;   // LDS
#define ASYNC_CP16(g, l)                                                     \
    __builtin_amdgcn_global_load_async_to_lds_b128((gv4i*)(g), (lv4i*)(l), 0, 0)
#define WAIT_ASYNC(n) __builtin_amdgcn_s_wait_asynccnt(n)
#else
#define ASYNC_CP16(g, l)                                                     \
    (*reinterpret_cast<float4*>(l) = *reinterpret_cast<const float4*>(g))
#define WAIT_ASYNC(n)
#endif

// ---------------------------------------------------------------------------
// WMMA helpers (V_WMMA_F32_16X16X4_F32, wave32)
// A: 16x4 fp32 -> 2 VGPR/lane: lane(0-15)=M, VGPR0/1 = K0/K1; lanes 16-31 = K2/K3
// B: 4x16 fp32 -> 2 VGPR/lane: lane&15 = N,  VGPR0/1 = K0/K1; lanes 16-31 = K2/K3
// C/D: 16x16 fp32, 8 VGPR: vgpr r -> row r (lanes 0-15) / row r+8 (lanes 16-31)
// ---------------------------------------------------------------------------
__device__ __forceinline__ v8f wmma4(v2f a, v2f b, v8f c) {
    return __builtin_amdgcn_wmma_f32_16x16x4_f32(
        false, a, false, b, (short)0, c, false, false);
}

__device__ __forceinline__ v2f load_a_frag(const float* src, int ld, int row, int k, int lane) {
    int m  = row + (lane & 15);
    int kk = k + ((lane >> 4) << 1);
    const float* p = src + m * ld + kk;   // contiguous pair -> ds_load_b64
    v2f a; a.x = p[0]; a.y = p[1];
    return a;
}

__device__ __forceinline__ v2f load_b_frag(const float* src, int ld, int k, int col, int lane) {
    int n  = col + (lane & 15);
    int kk = k + ((lane >> 4) << 1);
    v2f b;
    b.x = src[kk * ld + n];
    b.y = src[(kk + 1) * ld + n];
    return b;
}

// B operand taken as transpose of a row-major source: B[k][n] = src[n][k]
__device__ __forceinline__ v2f load_bt_frag(const float* src, int ld, int k, int col, int lane) {
    int n  = col + (lane & 15);
    int kk = k + ((lane >> 4) << 1);
    const float* p = src + n * ld + kk;   // contiguous pair -> ds_load_b64
    v2f b; b.x = p[0]; b.y = p[1];
    return b;
}

__device__ __forceinline__ v8f load_c_tile(const float* src, int ld, int row, int col, int lane) {
    int n  = col + (lane & 15);
    int m0 = row + ((lane >> 4) << 3);
    v8f c;
#pragma unroll
    for (int r = 0; r < 8; ++r) c[r] = src[(m0 + r) * ld + n];
    return c;
}

__device__ __forceinline__ void store_c_tile(float* dst, int ld, int row, int col, v8f c, int lane) {
    int n  = col + (lane & 15);
    int m0 = row + ((lane >> 4) << 3);
#pragma unroll
    for (int r = 0; r < 8; ++r) dst[(m0 + r) * ld + n] = c[r];
}

template <bool BT, bool NEGA>
__device__ __forceinline__ v8f tile_mm(const float* A, int lda, const float* Bm, int ldb,
                                       int row, int col, int kdim, v8f c, int lane) {
    for (int k = 0; k < kdim; k += 4) {
        v2f a = load_a_frag(A, lda, row, k, lane);
        if (NEGA) { a.x = -a.x; a.y = -a.y; }
        v2f b = BT ? load_bt_frag(Bm, ldb, k, col, lane)
                   : load_b_frag(Bm, ldb, k, col, lane);
        c = wmma4(a, b, c);
    }
    return c;
}

// ---------------------------------------------------------------------------
// LDS plan (floats), aliased across phases (61.0 KB total):
//   [    0.. 4096) F          (ph0-2)  -> HP(0..2048) + PHt(2048..4096) (ph3+)
//   [ 4096.. 8192) P_est      (ph0-1)  -> P_pred (ph2+)
//   [ 8192..12288) T=F*P      (ph1-2)  -> Aug[32x66](8192..10304) + K(10304..12352)
//   [12352..14400) H   (dedicated, async-prefetched)
//   [14400..15424) R   (dedicated, async-prefetched)
//   [15424..15616) vectors: x(64) xp(64) y(32) fbuf(32)
// ---------------------------------------------------------------------------
#define SMEM_FLOATS 15616

__global__ __launch_bounds__(256) void ekf_fused_kernel(
    const float* __restrict__ gx_est, const float* __restrict__ gP_est,
    const float* __restrict__ gF,     const float* __restrict__ gQ,
    const float* __restrict__ gz,     const float* __restrict__ gH,
    const float* __restrict__ gR,
    float* __restrict__ out_x, float* __restrict__ out_P, float* __restrict__ out_K)
{
    __shared__ float smem[SMEM_FLOATS];

    const int b    = blockIdx.x;
    const int tid  = threadIdx.x;
    const int lane = tid & 31;
    const int wave = tid >> 5;

    const float* Fg = gF    + (size_t)b * NDIM * NDIM;
    const float* Pg = gP_est+ (size_t)b * NDIM * NDIM;
    const float* Qg = gQ    + (size_t)b * NDIM * NDIM;
    const float* Hg = gH    + (size_t)b * MDIM * NDIM;
    const float* Rg = gR    + (size_t)b * MDIM * MDIM;
    const float* xg = gx_est+ (size_t)b * NDIM;
    const float* zg = gz    + (size_t)b * MDIM;

    float* sF   = smem;            // 64x64
    float* sHP  = smem;            // 32x64   (aliases dead F)
    float* sPHt = smem + 2048;     // 64x32   (aliases dead F)
    float* sP   = smem + 4096;     // 64x64 P_est -> P_pred
    float* sT   = smem + 8192;     // 64x64 temp
    float* sAug = smem + 8192;     // 32x66 augmented [S | I]  (over dead T)
    float* sK   = smem + 10304;    // 64x32                    (over dead T)
    float* sH   = smem + 12352;    // 32x64 dedicated
    float* sR   = smem + 14400;    // 32x32 dedicated
    float* sX   = smem + 15424;    // x_est[64]
    float* sXp  = smem + 15488;    // x_pred[64]
    float* sY   = smem + 15552;    // y[32]
    float* sFb  = smem + 15584;    // GJ factors[32]

    // ---- phase 0: async-stage F, P, H, R; regular load x --------------------
    // Per-wave async issue order/counts: F(4) P(4) H(2) R(1) = 11 total.
    {
#pragma unroll
        for (int i = 0; i < 4; ++i) {
            int idx = tid + i * 256;                       // 0..1023 float4s
            ASYNC_CP16(Fg + idx * 4, sF + idx * 4);
        }
#pragma unroll
        for (int i = 0; i < 4; ++i) {
            int idx = tid + i * 256;
            ASYNC_CP16(Pg + idx * 4, sP + idx * 4);
        }
#pragma unroll
        for (int i = 0; i < 2; ++i) {
            int idx = tid + i * 256;                       // 0..511 float4s
            ASYNC_CP16(Hg + idx * 4, sH + idx * 4);
        }
        ASYNC_CP16(Rg + tid * 4, sR + tid * 4);            // 256 float4s
        if (tid < NDIM) sX[tid] = xg[tid];
    }
    WAIT_ASYNC(3);            // F and P landed (H, R may still be in flight)
    __syncthreads();

    // ---- phase 1: T = F @ P_est  (16 tiles, 2/wave); x_pred = F @ x ---------
#pragma unroll
    for (int s = 0; s < 2; ++s) {
        int t = wave + s * 8;
        int row = (t >> 2) * 16, col = (t & 3) * 16;
        v8f c = {};
        c = tile_mm<false, false>(sF, NDIM, sP, NDIM, row, col, NDIM, c, lane);
        store_c_tile(sT, NDIM, row, col, c, lane);
    }
    if (tid < NDIM) {
        float s = 0.f;
#pragma unroll 8
        for (int i = 0; i < NDIM; ++i) s += sF[tid * NDIM + i] * sX[i];
        sXp[tid] = s;
    }
    __syncthreads();

    // ---- phase 2: P_pred = T @ F^T + Q  (C preloaded from global Q) ---------
#pragma unroll
    for (int s = 0; s < 2; ++s) {
        int t = wave + s * 8;
        int row = (t >> 2) * 16, col = (t & 3) * 16;
        v8f c = load_c_tile(Qg, NDIM, row, col, lane);
        c = tile_mm<true, false>(sT, NDIM, sF, NDIM, row, col, NDIM, c, lane);
        store_c_tile(sP, NDIM, row, col, c, lane);   // sP now holds P_pred
    }
    WAIT_ASYNC(1);            // H landed (R may still be in flight)
    __syncthreads();

    // ---- phase 3: HP = H @ P_pred (8 tiles); y = z - H x_pred; Aug |= I -----
    {
        int row = (wave >> 2) * 16, col = (wave & 3) * 16;
        v8f c = {};
        c = tile_mm<false, false>(sH, NDIM, sP, NDIM, row, col, NDIM, c, lane);
        store_c_tile(sHP, NDIM, row, col, c, lane);
    }
    if (tid < MDIM) {
        float s = 0.f;
#pragma unroll 8
        for (int i = 0; i < NDIM; ++i) s += sH[tid * NDIM + i] * sXp[i];
        sY[tid] = zg[tid] - s;
    }
#pragma unroll
    for (int e = 0; e < 4; ++e) {            // identity into Aug[:,32:64]
        int idx = tid + e * 256;             // 0..1023
        int r = idx >> 5, cc = idx & 31;
        sAug[r * 66 + 32 + cc] = (r == cc) ? 1.f : 0.f;
    }
    WAIT_ASYNC(0);            // R landed
    __syncthreads();

    // ---- phase 4: PHt = P_pred @ H^T (8 tiles); S = HP @ H^T + R + eps*I ----
    {
        int row = (wave >> 1) * 16, col = (wave & 1) * 16;
        v8f c = {};
        c = tile_mm<true, false>(sP, NDIM, sH, NDIM, row, col, NDIM, c, lane);
        store_c_tile(sPHt, MDIM, row, col, c, lane);
    }
    if (wave < 4) {
        int row = (wave >> 1) * 16, col = (wave & 1) * 16;
        v8f c = load_c_tile(sR, MDIM, row, col, lane);
        int n  = col + (lane & 15);
        int m0 = row + ((lane >> 4) << 3);
#pragma unroll
        for (int r = 0; r < 8; ++r) if (m0 + r == n) c[r] += EPSV;
        c = tile_mm<true, false>(sHP, NDIM, sH, NDIM, row, col, NDIM, c, lane);
        store_c_tile(sAug, 66, row, col, c, lane);
    }
    __syncthreads();

    // ---- phase 5: Gauss-Jordan inverse of S (32x32, augmented 32x66) --------
    for (int j = 0; j < MDIM; ++j) {
        float pinv = 1.0f / sAug[j * 66 + j];
        if (tid < MDIM && tid != j) sFb[tid] = sAug[tid * 66 + j];
        __syncthreads();
        if (tid < 64) sAug[j * 66 + tid] *= pinv;
        __syncthreads();
#pragma unroll
        for (int e = 0; e < 8; ++e) {
            int idx = tid + (e << 8);        // 0..2047
            int i = idx >> 6, cc = idx & 63;
            if (i != j) sAug[i * 66 + cc] -= sFb[i] * sAug[j * 66 + cc];
        }
        __syncthreads();
    }
    // S_inv now lives in sAug[:, 32:64]

    // ---- phase 6: K = PHt @ S_inv (8 tiles) ---------------------------------
    {
        int row = (wave >> 1) * 16, col = (wave & 1) * 16;
        v8f c = {};
        c = tile_mm<false, false>(sPHt, MDIM, sAug + 32, 66, row, col, MDIM, c, lane);
        store_c_tile(sK, MDIM, row, col, c, lane);
        store_c_tile(out_K + (size_t)b * NDIM * MDIM, MDIM, row, col, c, lane);
    }
    __syncthreads();

    // ---- phase 7: x_new = x_pred + K y ;  P_new = P_pred - K @ HP -----------
    if (tid < NDIM) {
        float s = sXp[tid];
#pragma unroll 8
        for (int j = 0; j < MDIM; ++j) s += sK[tid * MDIM + j] * sY[j];
        out_x[(size_t)b * NDIM + tid] = s;
    }
#pragma unroll
    for (int s = 0; s < 2; ++s) {
        int t = wave + s * 8;
        int row = (t >> 2) * 16, col = (t & 3) * 16;
        v8f c = load_c_tile(sP, NDIM, row, col, lane);       // P_pred tile
        c = tile_mm<false, true>(sK, MDIM, sHP, NDIM, row, col, MDIM, c, lane); // -K*HP
        store_c_tile(out_P + (size_t)b * NDIM * NDIM, NDIM, row, col, c, lane);
    }
}

extern "C" void kernel_launch(void* const* d_in, const int* in_sizes, int n_in,
                              void* d_out, int out_size, void* d_ws, size_t ws_size,
                              hipStream_t stream) {
    const float* x_est = (const float*)d_in[0];
    const float* P_est = (const float*)d_in[1];
    const float* F     = (const float*)d_in[2];
    const float* Q     = (const float*)d_in[3];
    const float* z     = (const float*)d_in[4];
    const float* H     = (const float*)d_in[5];
    const float* R     = (const float*)d_in[6];

    float* out   = (float*)d_out;
    float* out_x = out;                                       // [B,64,1]
    float* out_P = out_x + (size_t)BATCH * NDIM;              // [B,64,64]
    float* out_K = out_P + (size_t)BATCH * NDIM * NDIM;       // [B,64,32]

    ekf_fused_kernel<<<dim3(BATCH), dim3(256), 0, stream>>>(
        x_est, P_est, F, Q, z, H, R, out_x, out_P, out_K);
}